// LengthRegulator_53772990546269
// MI455X (gfx1250) — compile-verified
//
#include <hip/hip_runtime.h>

// ---------------------------------------------------------------------------
// LengthRegulator for MI455X (gfx1250).
// Memory-bound: ~201 MB of traffic -> ~8.7 us floor at 23.3 TB/s.
// No matrix math exists in this op, so WMMA is inapplicable; the CDNA5
// features that matter are async global->LDS staging (ASYNCcnt),
// global_prefetch_b8, and non-temporal streaming stores for the 189 MB output.
// ---------------------------------------------------------------------------

#define BB        16
#define CC        384
#define T_PHN     512
#define T_MAX     7680          // T_PHN * (MAX_DUR - 1)
#define T_MAX_PAD 8192          // padded so b128 tile loads never go OOB
#define CG        16            // channels per expand block
#define T_TILE    1024          // t positions per expand block (256 thr x 4)

typedef float f32x4 __attribute__((ext_vector_type(4)));

// Matches the builtin's parameter type from hipcc's diagnostic:
//   "__attribute__((__vector_size__(4 * sizeof(int)))) int __device__ *"
typedef int v4i __attribute__((vector_size(4 * sizeof(int))));
typedef __attribute__((address_space(1))) v4i gv4i;   // global (AS1 / __device__)
typedef __attribute__((address_space(3))) v4i lv4i;   // LDS    (AS3 / __shared__)

#if __has_builtin(__builtin_amdgcn_global_load_async_to_lds_b128)
#define HAVE_ASYNC_LDS 1
#else
#define HAVE_ASYNC_LDS 0
#endif

static __device__ __forceinline__ void wait_asynccnt0() {
#if __has_builtin(__builtin_amdgcn_s_wait_asynccnt)
    __builtin_amdgcn_s_wait_asynccnt(0);
#else
    asm volatile("s_wait_asynccnt 0x0" ::: "memory");
#endif
}

// ---------------------------------------------------------------------------
// Kernel 1: inclusive scan of durations per batch; emit cum[] and lengths.
// 16 blocks x 512 threads, LDS Hillis-Steele scan (9 steps).
// ---------------------------------------------------------------------------
__global__ void lr_cumsum(const int* __restrict__ ds,
                          int* __restrict__ cum,
                          float* __restrict__ len_out) {
    __shared__ int s[T_PHN];
    const int b = blockIdx.x;
    const int t = threadIdx.x;
    s[t] = ds[b * T_PHN + t];
    __syncthreads();
#pragma unroll
    for (int off = 1; off < T_PHN; off <<= 1) {
        const int v = (t >= off) ? s[t - off] : 0;
        __syncthreads();
        s[t] += v;
        __syncthreads();
    }
    cum[b * T_PHN + t] = s[t];
    if (t == T_PHN - 1) len_out[b] = (float)s[T_PHN - 1];
}

// ---------------------------------------------------------------------------
// Kernel 2: idx[b,t] = searchsorted_right(cum[b], t), or -1 for padded t.
// Binary search over the LDS-resident 512-entry cum row (9 iterations).
// Computed once per (b,t); reused by all 384 channels in kernel 3.
// ---------------------------------------------------------------------------
__global__ void lr_index(const int* __restrict__ cum,
                         int* __restrict__ idx) {
    __shared__ int s_cum[T_PHN];
    const int b = blockIdx.y;
    const int t = blockIdx.x * blockDim.x + threadIdx.x;   // [0, T_MAX_PAD)
    for (int i = threadIdx.x; i < T_PHN; i += blockDim.x)
        s_cum[i] = cum[b * T_PHN + i];
    __syncthreads();

    const int len = s_cum[T_PHN - 1];
    int r = -1;
    if (t < len && t < T_MAX) {
        int lo = 0, hi = T_PHN;                 // first j with cum[j] > t
#pragma unroll
        for (int it = 0; it < 9; ++it) {
            const int mid = (lo + hi) >> 1;
            if (s_cum[mid] <= t) lo = mid + 1; else hi = mid;
        }
        r = (lo < T_PHN - 1) ? lo : (T_PHN - 1);
    }
    idx[(size_t)b * T_MAX_PAD + t] = r;
}

// ---------------------------------------------------------------------------
// Kernel 3: the bandwidth kernel. One block = (b, 16-channel group, 1024-t
// tile). The 4 KB idx tile is staged into LDS with async global->LDS b128
// copies (ASYNCcnt), then each thread emits one non-temporal 16 B store per
// channel. xs rows (2 KB each, 12.6 MB total) stay L2/WGP$-resident; the
// next channel row is prefetched with global_prefetch_b8.
// ---------------------------------------------------------------------------
__global__ void lr_expand(const float* __restrict__ xs,
                          const int* __restrict__ idx,
                          float* __restrict__ out) {
    __shared__ int s_idx[T_TILE];
    const int tid = threadIdx.x;                // 0..255
    const int b  = blockIdx.z;
    const int c0 = blockIdx.y * CG;
    const int t0 = blockIdx.x * T_TILE;

    const int* gidx = idx + (size_t)b * T_MAX_PAD + t0;
#if HAVE_ASYNC_LDS
    __builtin_amdgcn_global_load_async_to_lds_b128(
        (gv4i*)(gidx + tid * 4),
        (lv4i*)(&s_idx[tid * 4]),
        /*offset=*/0, /*cpol=*/0);
    wait_asynccnt0();
#else
    ((int4*)s_idx)[tid] = ((const int4*)gidx)[tid];
#endif
    __syncthreads();

    const int tbase = t0 + tid * 4;
    const bool live = (tbase < T_MAX);          // T_MAX % 4 == 0, so all-or-nothing
    int i0 = -1, i1 = -1, i2 = -1, i3 = -1;
    if (live) {
        i0 = s_idx[tid * 4 + 0];
        i1 = s_idx[tid * 4 + 1];
        i2 = s_idx[tid * 4 + 2];
        i3 = s_idx[tid * 4 + 3];
    }

    const float* xrow = xs  + ((size_t)b * CC + c0) * (size_t)T_PHN;
    float*       orow = out + ((size_t)b * CC + c0) * (size_t)T_MAX + tbase;

#pragma unroll 4
    for (int cc = 0; cc < CG; ++cc) {
        if (cc + 1 < CG)
            __builtin_prefetch(xrow + T_PHN, 0, 1);   // next channel row -> global_prefetch_b8
        if (live) {
            f32x4 v;
            v.x = (i0 >= 0) ? xrow[i0] : 0.0f;
            v.y = (i1 >= 0) ? xrow[i1] : 0.0f;
            v.z = (i2 >= 0) ? xrow[i2] : 0.0f;
            v.w = (i3 >= 0) ? xrow[i3] : 0.0f;
            __builtin_nontemporal_store(v, (f32x4*)orow);  // streaming 189 MB output
        }
        xrow += T_PHN;
        orow += T_MAX;
    }
}

// ---------------------------------------------------------------------------
// Launch: inputs are {xs: f32[B,C,T_PHN], ds: i32[B,T_PHN], x_lengths: i32[B]}
// (x_lengths is unused by the reference). d_out = f32 out[B,C,T_MAX] followed
// by lengths[B] (written as float). Workspace: cum (32 KB) + idx (512 KB).
// ---------------------------------------------------------------------------
extern "C" void kernel_launch(void* const* d_in, const int* in_sizes, int n_in,
                              void* d_out, int out_size, void* d_ws, size_t ws_size,
                              hipStream_t stream) {
    (void)in_sizes; (void)n_in; (void)out_size; (void)ws_size;

    const float* xs = (const float*)d_in[0];
    const int*   ds = (const int*)d_in[1];
    // d_in[2] (x_lengths) intentionally unused, matching the reference.

    float* out     = (float*)d_out;
    float* len_out = out + (size_t)BB * CC * T_MAX;

    int* cum = (int*)d_ws;                 // BB * T_PHN ints
    int* idx = cum + BB * T_PHN;           // BB * T_MAX_PAD ints

    lr_cumsum<<<BB, T_PHN, 0, stream>>>(ds, cum, len_out);

    lr_index<<<dim3(T_MAX_PAD / 256, BB), 256, 0, stream>>>(cum, idx);

    lr_expand<<<dim3(T_MAX_PAD / T_TILE, CC / CG, BB), 256, 0, stream>>>(xs, idx, out);
}